// SpectrumMatchingLoss_8486855377163
// MI455X (gfx1250) — compile-verified
//
#include <hip/hip_runtime.h>

#define B_SZ 256
#define P_SZ 2048
#define L_SZ 64
#define V_SZ 28

#define PROTON 1.00727646688f
#define WATER  18.010564684f

typedef __attribute__((ext_vector_type(2))) float v2f;
typedef __attribute__((ext_vector_type(8))) float v8f;

// ---------------------------------------------------------------------------
// Kernel 1: predicted b/y ion masses per batch.
// One wave (32 lanes) per batch. exp_mass[b, l] = sum_v probs[b,l,v]*aa[v]
// computed with V_WMMA_F32_16X16X4_F32 (exact fp32): 4 M-tiles (L=64 rows)
// x 7 K-chunks (V=28). B-matrix = aa masses replicated across all 16 columns,
// so every column of D holds the dot product.
// A layout (ISA 7.12.2, 32-bit A 16x4): lanes 0-15 -> K=0 (VGPR0), K=1 (VGPR1);
// lanes 16-31 -> K=2, K=3. B uses the mirrored per-K striping.
// D layout (32-bit C/D 16x16): VGPR j holds row M=j (lanes 0-15) / M=8+j
// (lanes 16-31); columns are identical, so lanes 0 and 16 extract all 16 rows.
// ---------------------------------------------------------------------------
__global__ __launch_bounds__(32) void pred_masses_kernel(
    const float* __restrict__ probs,          // (B, L, V)
    const unsigned char* __restrict__ smask,  // (B, L) bool
    const float* __restrict__ aa,             // (V)
    float* __restrict__ pred)                 // (B, 2L) out
{
    const int b    = blockIdx.x;
    const int lane = threadIdx.x;
    const int half = lane >> 4;   // 0: K=0,1   1: K=2,3
    const int m15  = lane & 15;   // row within tile (A) / column (B,D)

    __shared__ float expm[L_SZ];
    __shared__ float prefix[L_SZ];

    const float* pb = probs + (size_t)b * L_SZ * V_SZ;

    for (int t = 0; t < 4; ++t) {
        v8f c = {};
        const int row = t * 16 + m15;
        #pragma unroll
        for (int k0 = 0; k0 < V_SZ; k0 += 4) {
            const int kk = k0 + half * 2;
            v2f a, bm;
            a.x  = pb[row * V_SZ + kk];
            a.y  = pb[row * V_SZ + kk + 1];
            bm.x = aa[kk];
            bm.y = aa[kk + 1];
            c = __builtin_amdgcn_wmma_f32_16x16x4_f32(
                    false, a, false, bm, (short)0, c, false, false);
        }
        // lanes 0 and 16 hold (identical-across-N) rows 0..7 / 8..15
        if (m15 == 0) {
            #pragma unroll
            for (int j = 0; j < 8; ++j) {
                const int l = t * 16 + half * 8 + j;
                const float msk = smask[b * L_SZ + l] ? 1.0f : 0.0f;
                expm[l] = c[j] * msk;
            }
        }
    }
    __syncthreads();

    if (lane == 0) {            // L=64 serial prefix sum: trivial cost
        float s = 0.0f;
        for (int l = 0; l < L_SZ; ++l) { s += expm[l]; prefix[l] = s; }
    }
    __syncthreads();

    const float total = prefix[L_SZ - 1];
    for (int l = lane; l < L_SZ; l += 32) {
        const float pf = prefix[l];
        pred[(size_t)b * (2 * L_SZ) + l]        = pf + PROTON;          // b-ion
        pred[(size_t)b * (2 * L_SZ) + L_SZ + l] = total - pf + WATER + PROTON; // y-ion
    }
}

// ---------------------------------------------------------------------------
// Kernel 2: per-batch weighted coverage.
// max_j exp(-0.5 (d_j/sigma)^2) == exp(-0.5 (min_j |d_j| / sigma)^2),
// so the 67M-pair inner loop is just v_sub + v_min(|x|) per pair, with one
// v_exp_f32 per peak. Predicted masses are staged in LDS.
// ---------------------------------------------------------------------------
__global__ __launch_bounds__(256) void coverage_kernel(
    const float* __restrict__ obs_m,          // (B, P)
    const float* __restrict__ obs_i,          // (B, P)
    const unsigned char* __restrict__ pmask,  // (B, P) bool
    const float* __restrict__ pred,           // (B, 2L)
    float* __restrict__ wc)                   // (B) out
{
    const int b   = blockIdx.x;
    const int tid = threadIdx.x;

    __shared__ float tj[2 * L_SZ];
    __shared__ float r1[256];
    __shared__ float r2[256];

    for (int j = tid; j < 2 * L_SZ; j += 256)
        tj[j] = pred[(size_t)b * (2 * L_SZ) + j];
    __syncthreads();

    const float cexp = -0.5f / (0.2f * 0.2f);  // -12.5
    float s1 = 0.0f, s2 = 0.0f;

    for (int p = tid; p < P_SZ; p += 256) {
        const size_t idx = (size_t)b * P_SZ + p;
        const float o = obs_m[idx];
        const float w = pmask[idx] ? obs_i[idx] : 0.0f;
        float m = 3.402823466e+38f;
        #pragma unroll
        for (int j = 0; j < 2 * L_SZ; ++j)
            m = fminf(m, __builtin_fabsf(o - tj[j]));
        s1 += w;
        s2 += w * __expf(cexp * m * m);
    }

    r1[tid] = s1; r2[tid] = s2;
    __syncthreads();
    for (int off = 128; off > 0; off >>= 1) {
        if (tid < off) { r1[tid] += r1[tid + off]; r2[tid] += r2[tid + off]; }
        __syncthreads();
    }
    if (tid == 0) wc[b] = r2[0] / fmaxf(r1[0], 1e-8f);
}

// ---------------------------------------------------------------------------
// Kernel 3: loss = 1 - mean_b(wc[b]); deterministic tree reduce.
// ---------------------------------------------------------------------------
__global__ __launch_bounds__(256) void finalize_kernel(
    const float* __restrict__ wc, float* __restrict__ out)
{
    __shared__ float r[256];
    const int tid = threadIdx.x;
    r[tid] = wc[tid];
    __syncthreads();
    for (int off = 128; off > 0; off >>= 1) {
        if (tid < off) r[tid] += r[tid + off];
        __syncthreads();
    }
    if (tid == 0) out[0] = 1.0f - r[0] * (1.0f / (float)B_SZ);
}

extern "C" void kernel_launch(void* const* d_in, const int* in_sizes, int n_in,
                              void* d_out, int out_size, void* d_ws, size_t ws_size,
                              hipStream_t stream) {
    const float*         probs = (const float*)d_in[0];         // (B,L,V)
    const float*         obs_m = (const float*)d_in[1];         // (B,P)
    const float*         obs_i = (const float*)d_in[2];         // (B,P)
    const unsigned char* pmask = (const unsigned char*)d_in[3]; // (B,P) bool
    const unsigned char* smask = (const unsigned char*)d_in[4]; // (B,L) bool
    const float*         aa    = (const float*)d_in[5];         // (V)

    float* pred = (float*)d_ws;                 // B * 2L floats = 128 KB
    float* wc   = pred + (size_t)B_SZ * 2 * L_SZ;

    pred_masses_kernel<<<B_SZ, 32, 0, stream>>>(probs, smask, aa, pred);
    coverage_kernel<<<B_SZ, 256, 0, stream>>>(obs_m, obs_i, pmask, pred, wc);
    finalize_kernel<<<1, 256, 0, stream>>>(wc, (float*)d_out);
}